// INRF_32332513804467
// MI455X (gfx1250) — compile-verified
//
#include <hip/hip_runtime.h>
#include <stdint.h>

// CDNA5 / gfx1250, wave32.
//  - Matrix op:  v_wmma_f32_16x16x4_f32 (native f32 WMMA)
//  - Staging:    global_load_async_to_lds_b32/b128 (ASYNCcnt path) via inline asm

typedef float v2f __attribute__((ext_vector_type(2)));
typedef float v8f __attribute__((ext_vector_type(8)));

#define HW        32
#define CC        16          // channels == filters
#define KTOT      144         // 3*3*16
#define KSTEPS    36          // KTOT / 4
#define PADW      34
#define ROWSTR    (PADW * CC) // 544 floats per padded row

// LDS partition (in floats)
#define OFF_IN    0            // 34*34*16 = 18496  (zero-padded input, one batch)
#define SZ_IN     18496
#define OFF_GP    18496        // 2304  (G[ij], pair-swizzled for b64 B-frag loads)
#define OFF_MR    20800        // 1024  (Msq[ij,:])
#define OFF_WR    21824        // 1024  (Wsq[ij,:])
#define OFF_OB    22848        // 64    (out accumulator, b*16+c)
#define SMEM_FLTS 22912
#define SMEM_BYTES (SMEM_FLTS * 4)

// LDS flat addresses truncate to the LDS byte offset (ISA §10.2), so the low
// 32 bits of a __shared__ pointer are a valid VDST for async-to-LDS.
__device__ __forceinline__ uint32_t lds_off(const void* p) {
    return (uint32_t)(uintptr_t)p;
}

// GLOBAL_LOAD_ASYNC_TO_LDS_*: per lane, copies from the global address in vaddr to the LDS address in vdst; ASYNCcnt-tracked.
__device__ __forceinline__ void async_ld_b32(uint32_t ldst, const void* gsrc) {
    asm volatile("global_load_async_to_lds_b32 %0, %1, off"
                 :: "v"(ldst), "v"(gsrc) : "memory");
}
__device__ __forceinline__ void async_ld_b128(uint32_t ldst, const void* gsrc) {
    asm volatile("global_load_async_to_lds_b128 %0, %1, off"
                 :: "v"(ldst), "v"(gsrc) : "memory");
}
__device__ __forceinline__ void wait_async0() {
    asm volatile("s_wait_asynccnt 0" ::: "memory");
}

__global__ __launch_bounds__(256) void INRF_32332513804467_kernel(
    const float* __restrict__ inp,  // (4, 32, 32, 16)
    const float* __restrict__ Mw,   // (1024, 1024)  Msq rows
    const float* __restrict__ Ww,   // (1024, 1024)  Wsq rows
    const float* __restrict__ G,    // (1024, 144, 16)  Gmat[k,f] per ij
    float* __restrict__ out)        // (4, 1024, 16)
{
    extern __shared__ float smem[];
    float* in_s = smem + OFF_IN;
    float* gp_s = smem + OFF_GP;
    float* mr_s = smem + OFF_MR;
    float* wr_s = smem + OFF_WR;
    float* ob_s = smem + OFF_OB;

    const int tid  = threadIdx.x;
    const int ij   = blockIdx.x;      // output spatial position, 0..1023
    const int lane = tid & 31;
    const int wave = tid >> 5;        // 0..7
    const int h    = lane >> 4;       // half-wave (selects K offset in A/B frags)
    const int n    = lane & 15;       // column index (channel f)

    // ---- zero padded input buffer (borders stay zero forever) + out acc ----
    for (int i = tid; i < SZ_IN; i += 256) in_s[i] = 0.0f;
    if (tid < 64) ob_s[tid] = 0.0f;

    // ---- async-stage G[ij] pair-swizzled: Gp[(k>>1)*32 + 2n + (k&1)] = G[ij,k,n]
    // (per-lane arbitrary LDS dest makes the swizzle free on the async path)
    {
        const float* gsrc = G + (size_t)ij * (KTOT * CC);
        #pragma unroll
        for (int s = 0; s < 9; ++s) {
            int idx = tid + 256 * s;                 // 0..2303
            int k = idx >> 4, nn = idx & 15;
            async_ld_b32(lds_off(gp_s + ((k >> 1) << 5) + (nn << 1) + (k & 1)),
                         gsrc + idx);
        }
    }
    // ---- async-stage M/W rows for this ij (contiguous -> b128, 1 per thread) ----
    {
        int f4 = tid << 2;                           // float index, 16B aligned
        async_ld_b128(lds_off(mr_s + f4), Mw + (size_t)ij * 1024 + f4);
        async_ld_b128(lds_off(wr_s + f4), Ww + (size_t)ij * 1024 + f4);
    }
    wait_async0();
    __syncthreads();

    for (int b = 0; b < 4; ++b) {
        // ---- async-stage inputs[b] into padded interior (y+1, x+1): b128 ----
        {
            const float* src = inp + (size_t)b * (HW * HW * CC);
            #pragma unroll
            for (int s = 0; s < 16; ++s) {
                int i4  = (tid + 256 * s) << 2;      // float index, 16B aligned
                int y   = i4 >> 9;                   // / (32*16)
                int rem = i4 & 511;                  // x*16 + c
                async_ld_b128(lds_off(in_s + (y + 1) * ROWSTR + CC + rem),
                              src + i4);
            }
        }
        wait_async0();
        __syncthreads();

        // ---- wave's 8 tiles: tile t = wave*8+tt -> y = t>>1, xhalf = t&1 ----
        // A-fragment base byte address (lane row m = lane&15 -> x = xh*16+m).
        // Padded address of (y+dy-1, x+dx-1, c) = y*544 + x*16 + (dy*544+dx*16+c).
        int baseA[8];
        #pragma unroll
        for (int tt = 0; tt < 8; ++tt) {
            int t  = wave * 8 + tt;
            int ty = t >> 1;
            int x  = ((t & 1) << 4) + (lane & 15);
            baseA[tt] = (ty * ROWSTR + x * CC) << 2;      // bytes
        }

        v8f acc[8];
        #pragma unroll
        for (int tt = 0; tt < 8; ++tt)
            acc[tt] = (v8f){0.f, 0.f, 0.f, 0.f, 0.f, 0.f, 0.f, 0.f};

        // ---- K loop: 36 steps of v_wmma_f32_16x16x4_f32 per tile ----
        for (int kk = 0; kk < KSTEPS; ++kk) {
            int k0 = (kk << 2) + (h << 1);                    // even; lanes 16-31 -> K+2
            int dy = (k0 >= 96) ? 2 : ((k0 >= 48) ? 1 : 0);
            int r  = k0 - dy * 48;
            int dx = r >> 4;
            int c  = r & 15;                                  // even -> b64 aligned
            int offA = (dy * ROWSTR + (dx << 4) + c) << 2;    // bytes

            // B fragment: (G[k0,n], G[k0+1,n]) — one b64 load, shared by 8 tiles
            v2f bfrag = *(const v2f*)((const char*)gp_s + ((((k0 >> 1) << 5) + (n << 1)) << 2));

            #pragma unroll
            for (int tt = 0; tt < 8; ++tt) {
                v2f afrag = *(const v2f*)((const char*)in_s + baseA[tt] + offA);
                // D = A(16x4 f32) * B(4x16 f32) + C  — v_wmma_f32_16x16x4_f32
                acc[tt] = __builtin_amdgcn_wmma_f32_16x16x4_f32(
                    false, afrag, false, bfrag, (short)0, acc[tt], false, false);
            }
        }

        // ---- fused epilogue: contrib = Msq*in - Wsq*relu(in - shifted) ----
        // D layout: lane (h,n), VGPR v -> row m = v + 8h, col n.
        float lane_acc = 0.0f;
        #pragma unroll
        for (int tt = 0; tt < 8; ++tt) {
            int t  = wave * 8 + tt;
            int ty = t >> 1;
            int xh = (t & 1) << 4;
            #pragma unroll
            for (int v = 0; v < 8; ++v) {
                int x    = xh + v + (h << 3);
                int yx   = ty * HW + x;
                float iv = in_s[(ty + 1) * ROWSTR + (x + 1) * CC + n];
                float d  = iv - acc[tt][v];
                d = fmaxf(d, 0.0f);                       // relu
                lane_acc += mr_s[yx] * iv - wr_s[yx] * d; // first - L*second (L=1)
            }
        }
        atomicAdd(&ob_s[b * 16 + n], lane_acc);           // ds_add_f32
        __syncthreads();   // protect in_s reuse + ob_s ordering
    }

    // ---- store out[b, ij, c] ----
    if (tid < 64) {
        int b = tid >> 4, c = tid & 15;
        out[(size_t)b * 16384 + (size_t)ij * 16 + c] = ob_s[tid];
    }
}

extern "C" void kernel_launch(void* const* d_in, const int* in_sizes, int n_in,
                              void* d_out, int out_size, void* d_ws, size_t ws_size,
                              hipStream_t stream) {
    (void)in_sizes; (void)n_in; (void)d_ws; (void)ws_size; (void)out_size;
    const float* inp = (const float*)d_in[0];   // inputs (4,32,32,16)
    const float* Mw  = (const float*)d_in[1];   // M (32,32,1,32,32,1) == (1024,1024)
    const float* Ww  = (const float*)d_in[2];   // W
    const float* G   = (const float*)d_in[3];   // G (32,32,3,3,16,16) == (1024,144,16)
    float* out = (float*)d_out;

    // >64KB dynamic LDS (CDNA5 allows up to 320KB/WG) — request it every call
    // (idempotent, deterministic; not a stream op so safe under graph capture).
    hipFuncSetAttribute((const void*)INRF_32332513804467_kernel,
                        hipFuncAttributeMaxDynamicSharedMemorySize, SMEM_BYTES);

    INRF_32332513804467_kernel<<<dim3(1024), dim3(256), SMEM_BYTES, stream>>>(
        inp, Mw, Ww, G, out);
}